// TSA_11699490915014
// MI455X (gfx1250) — compile-verified
//
#include <hip/hip_runtime.h>
#include <cmath>

typedef __attribute__((ext_vector_type(2))) float v2f;
typedef __attribute__((ext_vector_type(8))) float v8f;

#define B_  8
#define N_  400
#define T_  24
#define Y_  12
#define H_  64
#define D_  96
#define DL_ 64
#define TY_ (T_ * Y_)      // 288
#define BTY_ (B_ * TY_)    // 2304
#define NT_ 384            // threads per block (12 waves)
#define TC_ 6              // t-chunk per thread: 24 / (384/96)

__device__ __forceinline__ float fast_tanh(float x) {
#if __has_builtin(__builtin_amdgcn_tanhf)
    return __builtin_amdgcn_tanhf(x);
#else
    return tanhf(x);
#endif
}

__device__ __forceinline__ v8f wmma_f32_16x16x4(v2f a, v2f b, v8f c) {
    // D = A(16x4 f32) * B(4x16 f32) + C(16x16 f32)
    return __builtin_amdgcn_wmma_f32_16x16x4_f32(
        /*neg_a=*/false, a, /*neg_b=*/false, b,
        /*c_mod=*/(short)0, c, /*reuse_a=*/false, /*reuse_b=*/false);
}

// One workgroup per node n. 384 threads = 12 waves (wave32).
__global__ __launch_bounds__(NT_) void tsa_fused_kernel(
    const float* __restrict__ x_long,   // [B,N,T,Y]
    const float* __restrict__ x_short,  // [B,N,D]
    const float* __restrict__ Wq,       // [H,N,D]
    const float* __restrict__ Wk,       // [H,N,Y]
    const float* __restrict__ V,        // [H,N,Y]
    const float* __restrict__ encW,     // [N,Y,DL]
    const float* __restrict__ encb,     // [N,DL]
    float* __restrict__ out)            // [B,N,DL]
{
    __shared__ alignas(16) float  lds_wq[H_ * D_];    // 24576 B  Wq[h][d]
    __shared__ alignas(16) float  lds_xs[16 * D_];    //  6144 B  x_short padded to 16 rows
    __shared__ alignas(16) float  lds_Q [B_ * H_];    //  2048 B  Q[b][h]
    __shared__ alignas(16) float2 lds_vk[H_ * Y_];    //  6144 B  {Wk,V}[h][y] interleaved
    __shared__ alignas(16) float  lds_xl[BTY_];       //  9216 B  x_long[b][t][y]
    __shared__ alignas(16) float  lds_A [BTY_];       //  9216 B  logits -> exp weights
    __shared__ alignas(16) float  lds_xn[16 * 16];    //  1024 B  x_new padded (stride 16)
    __shared__ alignas(16) float  lds_ewT[DL_ * 16];  //  4096 B  enc_W transposed [d][y]

    const int n    = blockIdx.x;
    const int tid  = threadIdx.x;
    const int wave = tid >> 5;
    const int lane = tid & 31;

    // ---- Stage per-node slices into LDS ----
    // Wq slice: 6144 floats, vectorized float4 (rows are 384B = 16B-aligned).
    for (int i4 = tid; i4 < (H_ * D_) / 4; i4 += NT_) {
        const int i = i4 * 4;
        const int h = i / D_, d = i % D_;
        ((float4*)lds_wq)[i4] = *(const float4*)&Wq[(h * N_ + n) * D_ + d];
    }
    // x_short: rows 0..7 real (float4), rows 8..15 zero.
    for (int i4 = tid; i4 < (B_ * D_) / 4; i4 += NT_) {
        const int i = i4 * 4;
        const int b = i / D_, d = i % D_;
        ((float4*)lds_xs)[i4] = *(const float4*)&x_short[(b * N_ + n) * D_ + d];
    }
    for (int i = tid; i < B_ * D_; i += NT_) lds_xs[B_ * D_ + i] = 0.0f;
    // Wk/V interleaved as float2 {wk, v} per (h,y).
    for (int i = tid; i < H_ * Y_; i += NT_) {
        const int h = i / Y_, y = i % Y_;
        lds_vk[i] = make_float2(Wk[(h * N_ + n) * Y_ + y],
                                V [(h * N_ + n) * Y_ + y]);
    }
    // x_long tile: 2304 floats, float4 (tile base 1152B = 16B-aligned).
    for (int i4 = tid; i4 < BTY_ / 4; i4 += NT_) {
        const int i = i4 * 4;
        const int b = i / TY_, r = i % TY_;
        ((float4*)lds_xl)[i4] = *(const float4*)&x_long[(b * N_ + n) * TY_ + r];
    }
    // enc_W transposed to [d][y] with row stride 16 (so B-fragments are b64 loads).
    for (int i = tid; i < Y_ * DL_; i += NT_) {
        const int y = i / DL_, d = i % DL_;
        lds_ewT[d * 16 + y] = encW[n * Y_ * DL_ + i];
    }
    if (tid < 256) lds_xn[tid] = 0.0f;  // zero-pad x_new tile
    __syncthreads();

    // ---- Q[b,h] = sum_d x_short[b,d] * Wq[h,d]  via V_WMMA_F32_16X16X4_F32 ----
    // M=16 (B=8 zero-padded), N=16 (h-tile per wave 0..3), K=96 in 24 steps.
    if (wave < 4) {
        const int h0 = wave * 16;
        const int m  = lane & 15;          // A row / B column within tile
        const int ko = (lane >> 4) * 2;    // K sub-offset per lane half
        v8f c = {};
        #pragma unroll
        for (int kk = 0; kk < D_ / 4; ++kk) {
            const int k = kk * 4 + ko;
            const float2 af = *(const float2*)&lds_xs[m * D_ + k];          // b64
            const float2 bf = *(const float2*)&lds_wq[(h0 + m) * D_ + k];   // b64
            v2f a, b;
            a.x = af.x; a.y = af.y;
            b.x = bf.x; b.y = bf.y;
            c = wmma_f32_16x16x4(a, b, c);
        }
        if (lane < 16) {
            #pragma unroll
            for (int v = 0; v < B_; ++v)   // D row M=v is batch b=v
                lds_Q[v * H_ + h0 + lane] = c[v];
        }
    }
    __syncthreads();

    // ---- A[b,t,y] = sum_h V[h,y] * tanh(Q[b,h] + Wk[h,y] * x_long[b,t,y]) ----
    // Transcendental-bound hot loop. Each thread owns one (b,y) pair and a
    // 6-long t-chunk, so Q / {Wk,V} LDS fetches amortize over 12 tanh ops.
    {
        const int p  = tid % (B_ * Y_);       // (b,y) pair: 96 of them
        const int t0 = (tid / (B_ * Y_)) * TC_;  // t-chunk base: 0,6,12,18
        const int b  = p / Y_, y = p % Y_;
        float xl[TC_], acc[TC_];
        #pragma unroll
        for (int j = 0; j < TC_; ++j) {
            xl[j]  = lds_xl[b * TY_ + (t0 + j) * Y_ + y];
            acc[j] = 0.0f;
        }
        const float2* __restrict__ q2 = (const float2*)(lds_Q + b * H_);
        #pragma unroll 4
        for (int h2 = 0; h2 < H_ / 2; ++h2) {
            const float2 q  = q2[h2];                       // b64
            const float2 w0 = lds_vk[(2 * h2    ) * Y_ + y]; // b64
            const float2 w1 = lds_vk[(2 * h2 + 1) * Y_ + y]; // b64
            #pragma unroll
            for (int j = 0; j < TC_; ++j) {
                acc[j] += w0.y * fast_tanh(q.x + w0.x * xl[j]);
                acc[j] += w1.y * fast_tanh(q.y + w1.x * xl[j]);
            }
        }
        #pragma unroll
        for (int j = 0; j < TC_; ++j)
            lds_A[b * TY_ + (t0 + j) * Y_ + y] = acc[j];
    }
    __syncthreads();

    // ---- softmax over T and x_new[b,y] = sum_t Aw * x_long ----
    if (tid < B_ * Y_) {
        const int b = tid / Y_, y = tid % Y_;
        float mx = -INFINITY;
        #pragma unroll
        for (int t = 0; t < T_; ++t)
            mx = fmaxf(mx, lds_A[b * TY_ + t * Y_ + y]);
        float s = 0.0f;
        #pragma unroll
        for (int t = 0; t < T_; ++t) {
            const int idx = b * TY_ + t * Y_ + y;
            const float w = __expf(lds_A[idx] - mx);
            lds_A[idx] = w;            // only this thread touches these entries
            s += w;
        }
        float xn = 0.0f;
        #pragma unroll
        for (int t = 0; t < T_; ++t) {
            const int idx = b * TY_ + t * Y_ + y;
            xn += lds_A[idx] * lds_xl[idx];
        }
        lds_xn[b * 16 + y] = xn / s;
    }
    __syncthreads();

    // ---- out[b,d] = sum_y x_new[b,y] * enc_W[y,d] + enc_b[d]  via WMMA ----
    // M=16 (B=8 zero-padded), N=16 (d-tile per wave 0..3), K=12 in 3 steps.
    if (wave < 4) {
        const int d0 = wave * 16;
        const int m  = lane & 15;
        const int ko = (lane >> 4) * 2;
        v8f c = {};
        #pragma unroll
        for (int kk = 0; kk < 3; ++kk) {   // 3*4 = 12 = Y_ exactly
            const int k = kk * 4 + ko;
            const float2 af = *(const float2*)&lds_xn[m * 16 + k];            // b64
            const float2 bf = *(const float2*)&lds_ewT[(d0 + m) * 16 + k];    // b64
            v2f a, b;
            a.x = af.x; a.y = af.y;
            b.x = bf.x; b.y = bf.y;
            c = wmma_f32_16x16x4(a, b, c);
        }
        if (lane < 16) {
            #pragma unroll
            for (int v = 0; v < B_; ++v)
                out[(v * N_ + n) * DL_ + d0 + lane] =
                    c[v] + encb[n * DL_ + d0 + lane];
        }
    }
}

extern "C" void kernel_launch(void* const* d_in, const int* in_sizes, int n_in,
                              void* d_out, int out_size, void* d_ws, size_t ws_size,
                              hipStream_t stream) {
    const float* x_long  = (const float*)d_in[0];
    const float* x_short = (const float*)d_in[1];
    const float* Wq      = (const float*)d_in[2];
    const float* Wk      = (const float*)d_in[3];
    const float* V       = (const float*)d_in[4];
    const float* encW    = (const float*)d_in[5];
    const float* encb    = (const float*)d_in[6];
    float* out = (float*)d_out;

    tsa_fused_kernel<<<N_, NT_, 0, stream>>>(x_long, x_short, Wq, Wk, V,
                                             encW, encb, out);
}